// GraphGaze_88021059764679
// MI455X (gfx1250) — compile-verified
//
#include <hip/hip_runtime.h>
#include <hip/hip_bf16.h>

typedef __attribute__((ext_vector_type(2))) float v2f;
typedef __attribute__((ext_vector_type(8))) float v8f;

constexpr int Bc = 48, Pc = 56, Dc = 768, Hc = 8, Cc = 96;
constexpr int Nnodes = Bc * Pc;           // 2688
constexpr int Kdim = 768, Ncol = 768;
constexpr int TILES_N = Ncol / 16;        // 48
constexpr int TILES_M = Nnodes / 16;      // 168
constexpr float NEG_SLOPE = 0.2f;

// ---------------------------------------------------------------------------
// Generic fp32 WMMA GEMM: C[2688x768] = A[2688x768] @ B[768x768] + bias[768]
// One 16x16 output tile per wave; V_WMMA_F32_16X16X4_F32, K-loop unrolled x2
// with two accumulators to break the WMMA->WMMA RAW chain.
// ---------------------------------------------------------------------------
__global__ __launch_bounds__(256) void wmma_gemm_768(
    const float* __restrict__ A, const float* __restrict__ Bm,
    const float* __restrict__ bias, float* __restrict__ C) {
  const int wave = threadIdx.x >> 5;
  const int lane = threadIdx.x & 31;
  const int tile = blockIdx.x * 8 + wave;          // grid sized exactly
  const int tm = tile / TILES_N, tn = tile % TILES_N;
  const int kh  = lane >> 4;                       // 0: K pair {0,1}, 1: {2,3}
  const int l15 = lane & 15;

  const float* arow = A + (size_t)(tm * 16 + l15) * Kdim;
  const int bcol = tn * 16 + l15;

  v8f acc0 = {}; v8f acc1 = {};
  for (int k = 0; k < Kdim; k += 8) {
    const int ka0 = k + 2 * kh;
    v2f a0 = *(const v2f*)(arow + ka0);            // 8B aligned (k%4==0, 2*kh even)
    v2f b0; b0.x = Bm[(size_t)ka0 * Ncol + bcol];
            b0.y = Bm[(size_t)(ka0 + 1) * Ncol + bcol];
    acc0 = __builtin_amdgcn_wmma_f32_16x16x4_f32(false, a0, false, b0,
                                                 (short)0, acc0, false, false);
    const int ka1 = k + 4 + 2 * kh;
    v2f a1 = *(const v2f*)(arow + ka1);
    v2f b1; b1.x = Bm[(size_t)ka1 * Ncol + bcol];
            b1.y = Bm[(size_t)(ka1 + 1) * Ncol + bcol];
    acc1 = __builtin_amdgcn_wmma_f32_16x16x4_f32(false, a1, false, b1,
                                                 (short)0, acc1, false, false);
  }
  v8f acc = acc0 + acc1;

  const float bv = bias[bcol];
#pragma unroll
  for (int r = 0; r < 8; ++r) {
    const int row = tm * 16 + r + 8 * kh;          // C/D layout: lanes>=16 hold M+8
    C[(size_t)row * Ncol + bcol] = acc[r] + bv;
  }
}

// ---------------------------------------------------------------------------
// Dense per-(graph,head) GATv2 attention. v = nvp[b] <= 56, padded to 64.
// Logits + softmax on VALU, aggregation alpha(64x64) @ xl(64x96) via WMMA
// from LDS. Invalid nodes (self-loop only) pass xl through.
// ---------------------------------------------------------------------------
__global__ __launch_bounds__(256) void gatv2_attn(
    const float* __restrict__ XL, const float* __restrict__ XR,
    const float* __restrict__ att, const int* __restrict__ nvp,
    float* __restrict__ OUT) {
  __shared__ float s_xl[64][Cc];    // 24.0 KB  (rows >= v zeroed)
  __shared__ float s_al[64][65];    // 16.25 KB (logits -> alpha, padded stride)
  __shared__ float s_att[Cc];

  const int b = blockIdx.x, h = blockIdx.y;
  const int tid = threadIdx.x;
  const int v = nvp[b];
  const int g = Pc - v;                 // invalid (pass-through) nodes
  const int base = b * Pc + g;          // first valid node
  const size_t ho = (size_t)h * Cc;

  // pass-through for invalid nodes (alpha == 1 on the lone self-loop)
  for (int idx = tid; idx < g * Cc; idx += 256) {
    const int p = idx / Cc, c = idx % Cc;
    const size_t off = (size_t)(b * Pc + p) * Dc + ho + c;
    OUT[off] = XL[off];
  }
  if (tid < Cc) s_att[tid] = att[h * Cc + tid];
  for (int idx = tid; idx < 64 * Cc; idx += 256) {
    const int t = idx / Cc, c = idx % Cc;
    s_xl[t][c] = (t < v) ? XL[(size_t)(base + t) * Dc + ho + c] : 0.0f;
  }
  __syncthreads();

  // logits: thread handles i = tid/4, 16 j's; includes j==i (self-loop)
  {
    const int i = tid >> 2;
    const float* xr = XR + (size_t)(base + i) * Dc + ho;   // L2-resident
    for (int q = 0; q < 16; ++q) {
      const int j = (tid & 3) + 4 * q;
      float acc = 0.0f;
      if (i < v && j < v) {
        for (int c = 0; c < Cc; ++c) {
          float e = s_xl[j][c] + xr[c];
          e = (e > 0.0f) ? e : NEG_SLOPE * e;
          acc = fmaf(s_att[c], e, acc);
        }
      }
      s_al[i][j] = acc;
    }
  }
  __syncthreads();

  // row softmax over valid j; zero the padding so WMMA K-padding is inert
  if (tid < 64) {
    const int i = tid;
    if (i < v) {
      float mx = -1e30f;
      for (int j = 0; j < v; ++j) mx = fmaxf(mx, s_al[i][j]);
      float z = 0.0f;
      for (int j = 0; j < v; ++j) { float e = expf(s_al[i][j] - mx); s_al[i][j] = e; z += e; }
      const float rz = 1.0f / z;
      for (int j = 0; j < 64; ++j) s_al[i][j] = (j < v) ? s_al[i][j] * rz : 0.0f;
    } else {
      for (int j = 0; j < 64; ++j) s_al[i][j] = 0.0f;
    }
  }
  __syncthreads();

  // aggregation: OUT = alpha(64x64) @ s_xl(64x96); 24 tiles, 3 per wave
  const int wave = tid >> 5, lane = tid & 31;
  const int kh = lane >> 4, l15 = lane & 15;
  for (int t = wave; t < 24; t += 8) {             // uniform per wave: EXEC all-1s
    const int tm = t / 6, tn = t % 6;
    const int arow = tm * 16 + l15;
    const int bcol = tn * 16 + l15;
    v8f acc = {};
    for (int k = 0; k < 64; k += 4) {
      const int ka = k + 2 * kh;
      v2f a;  a.x  = s_al[arow][ka]; a.y  = s_al[arow][ka + 1];
      v2f bb; bb.x = s_xl[ka][bcol]; bb.y = s_xl[ka + 1][bcol];
      acc = __builtin_amdgcn_wmma_f32_16x16x4_f32(false, a, false, bb,
                                                  (short)0, acc, false, false);
    }
#pragma unroll
    for (int r = 0; r < 8; ++r) {
      const int row = tm * 16 + r + 8 * kh;
      if (row < v) OUT[(size_t)(base + row) * Dc + ho + bcol] = acc[r];
    }
  }
}

// ---------------------------------------------------------------------------
// bias_conv + LayerNorm(eps=1e-6) + exact GELU. One block per node.
// ---------------------------------------------------------------------------
__global__ __launch_bounds__(256) void ln_gelu(
    const float* __restrict__ IN, const float* __restrict__ bias,
    const float* __restrict__ gamma, const float* __restrict__ beta,
    float* __restrict__ ACT) {
  __shared__ float red[256];
  const int n = blockIdx.x, tid = threadIdx.x;
  const float* row = IN + (size_t)n * Dc;

  float vals[3]; float s = 0.0f;
#pragma unroll
  for (int q = 0; q < 3; ++q) { vals[q] = row[tid + 256 * q] + bias[tid + 256 * q]; s += vals[q]; }
  red[tid] = s; __syncthreads();
  for (int st = 128; st > 0; st >>= 1) { if (tid < st) red[tid] += red[tid + st]; __syncthreads(); }
  const float mu = red[0] * (1.0f / Dc);
  __syncthreads();

  float s2 = 0.0f;
#pragma unroll
  for (int q = 0; q < 3; ++q) { const float d = vals[q] - mu; s2 += d * d; }
  red[tid] = s2; __syncthreads();
  for (int st = 128; st > 0; st >>= 1) { if (tid < st) red[tid] += red[tid + st]; __syncthreads(); }
  const float rstd = rsqrtf(red[0] * (1.0f / Dc) + 1e-6f);

#pragma unroll
  for (int q = 0; q < 3; ++q) {
    const int c = tid + 256 * q;
    float y = (vals[q] - mu) * rstd * gamma[c] + beta[c];
    y = 0.5f * y * (1.0f + erff(y * 0.70710678118654752f));   // exact GELU
    ACT[(size_t)n * Dc + c] = y;
  }
}

// ---------------------------------------------------------------------------
extern "C" void kernel_launch(void* const* d_in, const int* in_sizes, int n_in,
                              void* d_out, int out_size, void* d_ws, size_t ws_size,
                              hipStream_t stream) {
  const float* x         = (const float*)d_in[0];
  const int*   nvp       = (const int*)  d_in[1];
  // d_in[2] = edge_index: structure reconstructed densely from nvp, unused
  const float* W_l       = (const float*)d_in[3];
  const float* b_l       = (const float*)d_in[4];
  const float* W_r       = (const float*)d_in[5];
  const float* b_r       = (const float*)d_in[6];
  const float* att       = (const float*)d_in[7];
  const float* bias_conv = (const float*)d_in[8];
  const float* ln_g      = (const float*)d_in[9];
  const float* ln_b      = (const float*)d_in[10];
  const float* W_ro      = (const float*)d_in[11];
  const float* b_ro      = (const float*)d_in[12];
  float* out = (float*)d_out;

  const size_t NT = (size_t)Nnodes * Dc;
  float* XL  = (float*)d_ws;
  float* XR  = XL + NT;
  float* PRE = XR + NT;
  float* ACT = XR;                 // XR dead after attention: reuse

  const int gemm_blocks = (TILES_M * TILES_N) / 8;   // 1008, exact

  wmma_gemm_768<<<gemm_blocks, 256, 0, stream>>>(x, W_l, b_l, XL);
  wmma_gemm_768<<<gemm_blocks, 256, 0, stream>>>(x, W_r, b_r, XR);
  gatv2_attn<<<dim3(Bc, Hc), 256, 0, stream>>>(XL, XR, att, nvp, PRE);
  ln_gelu<<<Nnodes, 256, 0, stream>>>(PRE, bias_conv, ln_g, ln_b, ACT);
  wmma_gemm_768<<<gemm_blocks, 256, 0, stream>>>(ACT, W_ro, b_ro, out);
}